// Nonstationary_26096221290913
// MI455X (gfx1250) — compile-verified
//
#include <hip/hip_runtime.h>
#include <math.h>

typedef __attribute__((ext_vector_type(2))) float v2f;
typedef __attribute__((ext_vector_type(8))) float v8f;

#define TN_PER_WAVE 4
#define WAVES_PER_BLOCK 8
#define LOG2E 1.44269504088896340736f

// ---------------------------------------------------------------------------
// Prep: per point i build
//   rowA[i] = ( x, y, ||x||^2, 1 )              (A-matrix row, K=4)
//   colB[j] = ( -2*x2, -2*y2, 1, ||x2||^2 )     (B-matrix col, K=4)
//   rowS[i] = 0.5*d1_i,  rowE[i] = exp(d1_i)
//   colS[j] = 0.5*d2_j,  colE[j] = exp(d2_j)
// so that A.B = sqdist and the epilogue is exp(rowS+colS - 2*sqdist/a)/a,
// a = 0.5*(rowE+colE).
// ---------------------------------------------------------------------------
__global__ void ps_prep_kernel(const float* __restrict__ X,
                               const float* __restrict__ X2,
                               float* __restrict__ rowA, float* __restrict__ colB,
                               float* __restrict__ rowS, float* __restrict__ rowE,
                               float* __restrict__ colS, float* __restrict__ colE,
                               int n1, int n2) {
    int i = blockIdx.x * blockDim.x + threadIdx.x;
    if (i < n1) {
        float x = X[2 * i + 0];
        float y = X[2 * i + 1];
        float nsq = x * x + y * y;
        float d = sqrtf(nsq);
        rowA[4 * i + 0] = x;
        rowA[4 * i + 1] = y;
        rowA[4 * i + 2] = nsq;
        rowA[4 * i + 3] = 1.0f;
        rowS[i] = 0.5f * d;
        rowE[i] = __builtin_amdgcn_exp2f(d * LOG2E);
    }
    if (i < n2) {
        float x = X2[2 * i + 0];
        float y = X2[2 * i + 1];
        float nsq = x * x + y * y;
        float d = sqrtf(nsq);
        colB[4 * i + 0] = -2.0f * x;
        colB[4 * i + 1] = -2.0f * y;
        colB[4 * i + 2] = 1.0f;
        colB[4 * i + 3] = nsq;
        colS[i] = 0.5f * d;
        colE[i] = __builtin_amdgcn_exp2f(d * LOG2E);
    }
}

// ---------------------------------------------------------------------------
// Main: one wave per 16-row strip, sweeping TN_PER_WAVE 16x16 tiles.
// sqdist tile via V_WMMA_F32_16X16X4_F32, epilogue via v_rcp_f32/v_exp_f32,
// NT stores for the 256 MB output (bigger than the 192 MB L2).
// ---------------------------------------------------------------------------
__global__ void __launch_bounds__(WAVES_PER_BLOCK * 32)
ps_cov_kernel(const float* __restrict__ rowA, const float* __restrict__ colB,
              const float* __restrict__ rowS, const float* __restrict__ rowE,
              const float* __restrict__ colS, const float* __restrict__ colE,
              float* __restrict__ out, int n2) {
    const int lane = threadIdx.x & 31;
    const int wave = threadIdx.x >> 5;
    const int l15  = lane & 15;
    const int half = lane >> 4;          // 0: lanes 0-15, 1: lanes 16-31
    const int kh   = half * 2;           // K offset {0,1} vs {2,3}

    const int tileM = blockIdx.y * WAVES_PER_BLOCK + wave;
    const int mbase = tileM * 16;

    // A fragment: lane holds row (mbase + l15), K = kh..kh+1  -> 2 VGPRs
    v2f afrag = *(const v2f*)(rowA + (size_t)(mbase + l15) * 4 + kh);

    // Row scalars for this lane's 8 output rows (M = half*8 + v)
    const int r0 = mbase + half * 8;
    float rs[8], re[8];
#pragma unroll
    for (int v = 0; v < 8; ++v) {
        rs[v] = rowS[r0 + v];
        re[v] = rowE[r0 + v];
    }

#pragma unroll
    for (int t = 0; t < TN_PER_WAVE; ++t) {
        const int tileN = blockIdx.x * TN_PER_WAVE + t;
        const int nc = tileN * 16 + l15;

        // B fragment: lane holds col nc, K = kh..kh+1 -> 2 VGPRs
        v2f bfrag = *(const v2f*)(colB + (size_t)nc * 4 + kh);
        const float cs = colS[nc];
        const float ce = colE[nc];

        // sqdist tile = A x B  (C initialized to 0, accumulated by WMMA)
        v8f q = {};
        q = __builtin_amdgcn_wmma_f32_16x16x4_f32(
                /*neg_a=*/false, afrag,
                /*neg_b=*/false, bfrag,
                /*c_mod=*/(short)0, q,
                /*reuse_a=*/false, /*reuse_b=*/false);

        // Epilogue: out = exp(rs + cs - 2*q/a) / a, a = 0.5*(re + ce)
#pragma unroll
        for (int v = 0; v < 8; ++v) {
            float a   = 0.5f * (re[v] + ce);
            float r   = __builtin_amdgcn_rcpf(a);       // a >= 1, safe
            float arg = rs[v] + cs - 2.0f * q[v] * r;
            float val = __builtin_amdgcn_exp2f(arg * LOG2E) * r;
            __builtin_nontemporal_store(val, out + (size_t)(r0 + v) * n2 + nc);
        }
    }
}

extern "C" void kernel_launch(void* const* d_in, const int* in_sizes, int n_in,
                              void* d_out, int out_size, void* d_ws, size_t ws_size,
                              hipStream_t stream) {
    const float* X  = (const float*)d_in[0];   // [n1,2] f32
    const float* X2 = (const float*)d_in[1];   // [n2,2] f32
    float* out = (float*)d_out;

    const int n1 = in_sizes[0] / 2;            // 8192
    const int n2 = in_sizes[1] / 2;            // 8192

    // Workspace layout (floats)
    float* w    = (float*)d_ws;
    float* rowA = w;                    // 4*n1
    float* colB = rowA + (size_t)4 * n1; // 4*n2
    float* rowS = colB + (size_t)4 * n2; // n1
    float* rowE = rowS + n1;             // n1
    float* colS = rowE + n1;             // n2
    float* colE = colS + n2;             // n2

    {
        int nmax = n1 > n2 ? n1 : n2;
        int threads = 256;
        int blocks = (nmax + threads - 1) / threads;
        ps_prep_kernel<<<blocks, threads, 0, stream>>>(X, X2, rowA, colB,
                                                       rowS, rowE, colS, colE,
                                                       n1, n2);
    }

    {
        const int tilesM = n1 / 16;                       // 512
        const int tilesN = n2 / 16;                       // 512
        dim3 grid(tilesN / TN_PER_WAVE,                   // 128
                  tilesM / WAVES_PER_BLOCK);              // 64
        dim3 block(WAVES_PER_BLOCK * 32);                 // 256 threads = 8 waves
        ps_cov_kernel<<<grid, block, 0, stream>>>(rowA, colB, rowS, rowE,
                                                  colS, colE, out, n2);
    }
}